// NN_log_den_t_27281632264838
// MI455X (gfx1250) — compile-verified
//
#include <hip/hip_runtime.h>
#include <hip/hip_bf16.h>

// D = A(16x4 f32) * B(4x16 f32) + C(16x16 f32), wave32, fp32-exact WMMA.
typedef __attribute__((ext_vector_type(2))) float v2f;
typedef __attribute__((ext_vector_type(8))) float v8f;

#define N_ROWS 8192
#define D_COLS 4096
#define WAVES_PER_BLOCK 16
#define BLOCK_THREADS (WAVES_PER_BLOCK * 32)          // 512
#define COLS_PER_WAVE (D_COLS / WAVES_PER_BLOCK)      // 256

__global__ __launch_bounds__(BLOCK_THREADS) void
NN_log_den_kernel(const float* __restrict__ X,
                  const float* __restrict__ mean,
                  const float* __restrict__ df,
                  float* __restrict__ out) {
    const int tid   = threadIdx.x;
    const int lane  = tid & 31;        // wave32
    const int wave  = tid >> 5;        // 0..15
    const int mrow  = lane & 15;       // A-matrix row M for this lane
    const int khalf = lane >> 4;       // 0 -> K={0,1}, 1 -> K={2,3}

    const int row = blockIdx.x * 16 + mrow;
    const float* __restrict__ xrow = X + (size_t)row * D_COLS;

    // First column this lane touches within its wave's 256-col slice.
    const int cbase = wave * COLS_PER_WAVE + khalf * 2;

    v8f c = {0.f, 0.f, 0.f, 0.f, 0.f, 0.f, 0.f, 0.f};
    const v2f bones = {1.0f, 1.0f};    // all-ones B: D[m][n] = sum_k A[m][k]

    #pragma unroll 8
    for (int k = 0; k < COLS_PER_WAVE; k += 4) {
        const float2 xv = *(const float2*)(xrow + cbase + k);
        const float2 mv = *(const float2*)(mean + cbase + k);
        const float d0 = xv.x - mv.x;
        const float d1 = xv.y - mv.y;
        v2f a;
        a.x = d0 * d0;
        a.y = d1 * d1;
        // (neg_a, A, neg_b, B, c_mod, C, reuse_a, reuse_b)
        c = __builtin_amdgcn_wmma_f32_16x16x4_f32(
                false, a, false, bones, (short)0, c, false, false);
    }

    // C layout: VGPR r, lanes 0-15 -> M=r; lanes 16-31 -> M=8+r (all N equal).
    __shared__ float red[WAVES_PER_BLOCK * 16];
    if (lane == 0) {
        #pragma unroll
        for (int r = 0; r < 8; ++r) red[wave * 16 + r] = c[r];
    } else if (lane == 16) {
        #pragma unroll
        for (int r = 0; r < 8; ++r) red[wave * 16 + 8 + r] = c[r];
    }
    __syncthreads();

    if (tid < 16) {
        float s = 0.f;
        #pragma unroll
        for (int w = 0; w < WAVES_PER_BLOCK; ++w) s += red[w * 16 + tid];
        const float dfv = df[0];
        const float v = -0.5f * (dfv + (float)D_COLS) * logf(1.0f + s / dfv);
        out[blockIdx.x * 16 + tid] = v;
    }
}

extern "C" void kernel_launch(void* const* d_in, const int* in_sizes, int n_in,
                              void* d_out, int out_size, void* d_ws, size_t ws_size,
                              hipStream_t stream) {
    const float* X    = (const float*)d_in[0];
    const float* mean = (const float*)d_in[1];
    const float* df   = (const float*)d_in[2];
    float* out        = (float*)d_out;

    dim3 grid(N_ROWS / 16);            // 512 workgroups x 16 rows
    dim3 block(BLOCK_THREADS);         // 16 waves (wave32) -> 8192 waves total
    hipLaunchKernelGGL(NN_log_den_kernel, grid, block, 0, stream,
                       X, mean, df, out);
}